// AperiodicKNN_PyG_76347338654307
// MI455X (gfx1250) — compile-verified
//
#include <hip/hip_runtime.h>
#include <hip/hip_bf16.h>
#include <math.h>

typedef __attribute__((ext_vector_type(2))) float v2f;
typedef __attribute__((ext_vector_type(8))) float v8f;

#define KNN_K 19
#define TILE 16
#define WAVES_PER_BLOCK 8
#define THREADS (WAVES_PER_BLOCK * 32)

// Fused kNN: score tiles via V_WMMA_F32_16X16X4_F32 with the distance epilogue
// folded into the K=4 slot:  A_j = (x,y,z,|p_j|^2),  B_i = (-2x,-2y,-2z,1)
//  => D[m][n] = |p_j|^2 - 2 p_i.p_j   (rank-equivalent to d^2 per center)
__global__ __launch_bounds__(THREADS) void knn_wmma_kernel(
    const float* __restrict__ pos, int n,
    int* __restrict__ out_idx, float* __restrict__ out_dist, int nk)
{
  __shared__ float ldsS[WAVES_PER_BLOCK * 32 * KNN_K];
  __shared__ int   ldsI[WAVES_PER_BLOCK * 32 * KNN_K];

  const int lane = threadIdx.x & 31;
  const int wave = threadIdx.x >> 5;
  const int n16  = lane & 15;
  const int hi   = lane >> 4;                 // 0: holds K=0,1  1: holds K=2,3
  const int ibase = (blockIdx.x * WAVES_PER_BLOCK + wave) * TILE;
  const int i = ibase + n16;                  // center owned by this lane (column n)

  // ---- B operand (centers), built once:  4x16 (KxN), column n = center i
  // lanes 0-15: {K0,K1} = {-2x_i,-2y_i}; lanes 16-31: {K2,K3} = {-2z_i, 1}
  const float xi = pos[3*i+0], yi = pos[3*i+1], zi = pos[3*i+2];
  v2f b;
  b.x = hi ? (-2.0f * zi) : (-2.0f * xi);
  b.y = hi ? 1.0f         : (-2.0f * yi);
  const float sqi = xi*xi + yi*yi + zi*zi;

  // ---- per-lane sorted (ascending) top-K list, register resident
  float d[KNN_K]; int id[KNN_K];
#pragma unroll
  for (int t = 0; t < KNN_K; ++t) { d[t] = 3.4e38f; id[t] = 0; }

#pragma unroll 2
  for (int jb = 0; jb < n; jb += TILE) {
    // A operand: candidate tile, row m = lane%16, point j = jb+m
    // lanes 0-15: {x_j, y_j}; lanes 16-31: {z_j, |p_j|^2}
    const int jA = jb + n16;
    const float xj = pos[3*jA+0], yj = pos[3*jA+1], zj = pos[3*jA+2];
    const float sqj = xj*xj + yj*yj + zj*zj;
    v2f a;
    a.x = hi ? zj  : xj;
    a.y = hi ? sqj : yj;

    v8f c = {};
    c = __builtin_amdgcn_wmma_f32_16x16x4_f32(false, a, false, b,
                                              (short)0, c, false, false);

    // lane holds D[m][n] for m = v + 8*hi at vector slot v; n = this center
    const int j0 = jb + 8 * hi;
#pragma unroll
    for (int v = 0; v < 8; ++v) {
      const float s = c[v];
      const int jj = j0 + v;
      if (jj != i && s < d[KNN_K-1]) {        // self excluded; rare insert path
        d[KNN_K-1] = s; id[KNN_K-1] = jj;
#pragma unroll
        for (int t = KNN_K-1; t > 0; --t) {   // one bubble pass restores order
          if (d[t] < d[t-1]) {
            float td = d[t]; d[t] = d[t-1]; d[t-1] = td;
            int   ti = id[t]; id[t] = id[t-1]; id[t-1] = ti;
          }
        }
      }
    }
  }

  // ---- dump both half-lists, merge lane n with lane n+16
  const int slot = (wave * 32 + lane) * KNN_K;
#pragma unroll
  for (int t = 0; t < KNN_K; ++t) { ldsS[slot + t] = d[t]; ldsI[slot + t] = id[t]; }
  __syncthreads();

  if (lane < 16) {
    const int b0 = (wave * 32 + lane) * KNN_K;
    const int b1 = (wave * 32 + lane + 16) * KNN_K;
    int p0 = 0, p1 = 0;
    const int e = i * KNN_K;
    for (int r = 0; r < KNN_K; ++r) {
      const float s0 = ldsS[b0 + p0], s1 = ldsS[b1 + p1];
      const bool t0 = s0 <= s1;
      const float s = t0 ? s0 : s1;
      const int  jj = t0 ? ldsI[b0 + p0] : ldsI[b1 + p1];
      p0 += (int)t0; p1 += (int)!t0;
      float d2 = s + sqi;                     // restore true squared distance
      d2 = d2 > 0.0f ? d2 : 0.0f;
      out_idx[e + r]      = jj;               // edge_index row 0: neighbor
      out_idx[nk + e + r] = i;                // edge_index row 1: center
      out_dist[e + r]     = sqrtf(d2);
    }
  }
}

extern "C" void kernel_launch(void* const* d_in, const int* in_sizes, int n_in,
                              void* d_out, int out_size, void* d_ws, size_t ws_size,
                              hipStream_t stream) {
  const float* pos = (const float*)d_in[0];
  const int n  = in_sizes[0] / 3;             // 16384
  const int nk = n * KNN_K;
  int*   out_idx  = (int*)d_out;                          // [2, n*k] int32
  float* out_dist = (float*)d_out + 2 * (size_t)nk;       // [n*k] float32
  const int blocks = n / (TILE * WAVES_PER_BLOCK);
  knn_wmma_kernel<<<blocks, THREADS, 0, stream>>>(pos, n, out_idx, out_dist, nk);
}